// Axon_53489522704543
// MI455X (gfx1250) — compile-verified
//
#include <hip/hip_runtime.h>
#include <stdint.h>

#define NUM_SOURCE   16384
#define NUM_TARGET   16384
#define NUM_BRANCHES 64
#define BATCH        32

#define BPB     2                              // batches privatized per workgroup
#define CHUNKS  16                             // source chunks per batch-pair
#define SRC_PER_WG   (NUM_SOURCE / CHUNKS)     // 1024
#define EDGES_PER_WG (SRC_PER_WG * NUM_BRANCHES) // 65536
#define THREADS 256

// log2(0.9): decay = 0.9^d = exp2(d * log2(0.9)), d in [0,6)
#define LOG2_SMOOTH (-0.15200309344504997f)

__global__ void axon_zero_kernel(float* __restrict__ out, int n4) {
    int i = blockIdx.x * blockDim.x + threadIdx.x;
    if (i < n4) {
        reinterpret_cast<float4*>(out)[i] = make_float4(0.f, 0.f, 0.f, 0.f);
    }
}

// CDNA5 async copy: 16B global -> LDS, tracked by ASYNCcnt (no VGPR data path).
__device__ __forceinline__ void async_f4_to_lds(uint32_t lds_byte_off, const float* g) {
    asm volatile("global_load_async_to_lds_b128 %0, %1, off"
                 :
                 : "v"(lds_byte_off), "v"(g)
                 : "memory");
}

__device__ __forceinline__ void wait_asynccnt0() {
    asm volatile("s_wait_asynccnt 0" ::: "memory");
}

__launch_bounds__(THREADS)
__global__ void axon_scatter_kernel(const float* __restrict__ spikes,       // [B, S]
                                    const float* __restrict__ attenuation,  // [S, Br]
                                    const int*   __restrict__ targets,      // [S, Br]
                                    const int*   __restrict__ delays,       // [S, Br]
                                    float*       __restrict__ out)          // [B, T]
{
    __shared__ float hist[BPB * NUM_TARGET];   // 128 KB privatized histograms
    __shared__ float spk[BPB * SRC_PER_WG];    // 8 KB spike slices

    const int tid = threadIdx.x;
    const int s0  = blockIdx.x * SRC_PER_WG;   // first source of this chunk
    const int b0  = blockIdx.y * BPB;          // first batch of this pair

    // --- 1) launch async spike-slice copy into LDS (BPB * 256 * 16B) ------
    const uint32_t spk_base = (uint32_t)(uintptr_t)&spk[0];
#pragma unroll
    for (int b = 0; b < BPB; ++b) {
        const float* g = spikes + (size_t)(b0 + b) * NUM_SOURCE + s0 + tid * 4;
        async_f4_to_lds(spk_base + (uint32_t)(b * SRC_PER_WG + tid * 4) * 4u, g);
    }

    // --- 2) zero histograms with b128 LDS stores while copy is in flight --
    {
        float4  z  = make_float4(0.f, 0.f, 0.f, 0.f);
        float4* h4 = reinterpret_cast<float4*>(hist);
        for (int i = tid; i < (BPB * NUM_TARGET) / 4; i += THREADS)
            h4[i] = z;
    }

    wait_asynccnt0();
    __syncthreads();

    // --- 3) stream edges (b128 loads), scatter into LDS via ds_add_f32 ----
    const size_t edge_base = (size_t)s0 * NUM_BRANCHES;  // 16B aligned (mult of 65536)
    const float4* __restrict__ attv = reinterpret_cast<const float4*>(attenuation + edge_base);
    const int4*   __restrict__ tgtv = reinterpret_cast<const int4*>(targets + edge_base);
    const int4*   __restrict__ dlyv = reinterpret_cast<const int4*>(delays + edge_base);

    for (int i = tid; i < EDGES_PER_WG / 4; i += THREADS) {
        float4 a = attv[i];
        int4   t = tgtv[i];
        int4   d = dlyv[i];

        // 4 consecutive edges share one source (64 branches, quad-aligned)
        const int   sl  = (i * 4) >> 6;
        const float sp0 = spk[sl];
        const float sp1 = spk[SRC_PER_WG + sl];

        const float w0 = fminf(fmaxf(a.x, 0.f), 1.f) * exp2f((float)d.x * LOG2_SMOOTH);
        const float w1 = fminf(fmaxf(a.y, 0.f), 1.f) * exp2f((float)d.y * LOG2_SMOOTH);
        const float w2 = fminf(fmaxf(a.z, 0.f), 1.f) * exp2f((float)d.z * LOG2_SMOOTH);
        const float w3 = fminf(fmaxf(a.w, 0.f), 1.f) * exp2f((float)d.w * LOG2_SMOOTH);

        atomicAdd(&hist[t.x], sp0 * w0);
        atomicAdd(&hist[t.y], sp0 * w1);
        atomicAdd(&hist[t.z], sp0 * w2);
        atomicAdd(&hist[t.w], sp0 * w3);

        atomicAdd(&hist[NUM_TARGET + t.x], sp1 * w0);
        atomicAdd(&hist[NUM_TARGET + t.y], sp1 * w1);
        atomicAdd(&hist[NUM_TARGET + t.z], sp1 * w2);
        atomicAdd(&hist[NUM_TARGET + t.w], sp1 * w3);
    }

    __syncthreads();

    // --- 4) flush: sequential, conflict-free global_atomic_add_f32 --------
    float* __restrict__ out0 = out + (size_t)b0 * NUM_TARGET;
    for (int i = tid; i < NUM_TARGET; i += THREADS) {
        atomicAdd(&out0[i],              hist[i]);
        atomicAdd(&out0[NUM_TARGET + i], hist[NUM_TARGET + i]);
    }
}

extern "C" void kernel_launch(void* const* d_in, const int* in_sizes, int n_in,
                              void* d_out, int out_size, void* d_ws, size_t ws_size,
                              hipStream_t stream) {
    const float* spikes = (const float*)d_in[0];   // [32, 16384] f32
    const float* att    = (const float*)d_in[1];   // [16384, 64] f32
    const int*   tgt    = (const int*)d_in[2];     // [16384, 64] i32
    const int*   dly    = (const int*)d_in[3];     // [16384, 64] i32
    float*       out    = (float*)d_out;           // [32, 16384] f32

    // zero the (poisoned) output, then scatter-accumulate into it
    const int n4 = out_size / 4;  // 131072 float4s
    axon_zero_kernel<<<(n4 + 255) / 256, 256, 0, stream>>>(out, n4);

    dim3 grid(CHUNKS, BATCH / BPB);  // 16 x 16 = 256 workgroups
    axon_scatter_kernel<<<grid, THREADS, 0, stream>>>(spikes, att, tgt, dly, out);
}